// GNN_15221364097129
// MI455X (gfx1250) — compile-verified
//
#include <hip/hip_runtime.h>

#define NNODES 100000
#define NEDGES 1600000

typedef __attribute__((ext_vector_type(2))) float v2f;
typedef __attribute__((ext_vector_type(8))) float v8f;

// ---------------- utility kernels ----------------

__global__ void zero_f32(float* __restrict__ p, long n) {
  long i = (long)blockIdx.x * blockDim.x + threadIdx.x;
  long stride = (long)gridDim.x * blockDim.x;
  for (; i < n; i += stride) p[i] = 0.0f;
}

__global__ void deg_accum(const int* __restrict__ dst, const float* __restrict__ w,
                          float* __restrict__ deg, int E) {
  int e = blockIdx.x * blockDim.x + threadIdx.x;
  if (e < E) unsafeAtomicAdd(&deg[dst[e]], w[e]);
}

__global__ void dinv_selfw(const float* __restrict__ deg, float* __restrict__ dinv,
                           float* __restrict__ selfw, int n) {
  int i = blockIdx.x * blockDim.x + threadIdx.x;
  if (i < n) {
    float d = deg[i] + 1.0f;          // self-loop weight 1 included in degree
    float r = rsqrtf(d);              // d > 0 always
    dinv[i]  = r;
    selfw[i] = r * r;                 // norm of the self-loop edge
  }
}

__global__ void edge_norm(const int* __restrict__ src, const int* __restrict__ dst,
                          const float* __restrict__ w, const float* __restrict__ dinv,
                          float* __restrict__ norm, int E) {
  int e = blockIdx.x * blockDim.x + threadIdx.x;
  if (e < E) norm[e] = dinv[src[e]] * w[e] * dinv[dst[e]];
}

// ---------------- WMMA f32 GEMM: H[M x NC] = A[M x K] * W[K x NC] ----------------
// One wave per 16x16 output tile; K-loop fully unrolled -> 32x v_wmma_f32_16x16x4_f32.
// A frag (16x4):  v0 = K0(l0-15)/K2(l16-31), v1 = K1/K3
// B frag (4x16):  v0 = K0(l0-15)/K2(l16-31), v1 = K1/K3  (col = lane&15)
// C/D (16x16):    vgpr r: M=r (lanes 0-15), M=r+8 (lanes 16-31)

template <int K, int NC>
__global__ void gemm_f32_wmma(const float* __restrict__ A, const float* __restrict__ W,
                              float* __restrict__ H, int M) {
  const int lane = threadIdx.x & 31;
  const int wave = (blockIdx.x * blockDim.x + threadIdx.x) >> 5;
  const int tpr  = NC >> 4;               // tiles per row
  const int mt   = wave / tpr;
  const int nt   = wave % tpr;
  if (mt * 16 >= M) return;               // wave-uniform; EXEC stays all-ones

  const int half = lane >> 4;             // 0 or 1
  const int l15  = lane & 15;

  const float* arow = A + (size_t)(mt * 16 + l15) * K + 2 * half;
  const float* bcol = W + (size_t)(2 * half) * NC + nt * 16 + l15;

  v8f c = {};
#pragma unroll
  for (int k = 0; k < K; k += 4) {
    v2f a = *(const v2f*)(arow + k);      // A[m][k+2h], A[m][k+2h+1]
    v2f b;
    b.x = bcol[(size_t)k * NC];           // W[k+2h  ][n]
    b.y = bcol[(size_t)(k + 1) * NC];     // W[k+2h+1][n]
    c = __builtin_amdgcn_wmma_f32_16x16x4_f32(
        /*neg_a=*/false, a, /*neg_b=*/false, b,
        /*c_mod=*/(short)0, c, /*reuse_a=*/false, /*reuse_b=*/false);
  }

  float* out = H + (size_t)(mt * 16 + 8 * half) * NC + nt * 16 + l15;
#pragma unroll
  for (int r = 0; r < 8; ++r)
    out[(size_t)r * NC] = c[r];
}

// ---------------- edge scatter: agg[dst] += h[src] * norm ----------------
// One wave per edge; each lane moves one float4 chunk (D=128 -> 32 lanes, D=64 -> 16).

template <int D>
__global__ void scatter_edges(const float* __restrict__ h, const int* __restrict__ src,
                              const int* __restrict__ dst, const float* __restrict__ norm,
                              float* __restrict__ agg, int E) {
  const int wave = (blockIdx.x * blockDim.x + threadIdx.x) >> 5;
  const int lane = threadIdx.x & 31;
  if (wave >= E) return;
  const int   s  = src[wave];
  const int   d  = dst[wave];
  const float nw = norm[wave];
  constexpr int NVEC = D >> 2;            // float4 chunks per row
  if (lane < NVEC) {
    const float4* hp = (const float4*)(h + (size_t)s * D);
    float4 v = hp[lane];
    float* ap = agg + (size_t)d * D + lane * 4;
    unsafeAtomicAdd(ap + 0, v.x * nw);
    unsafeAtomicAdd(ap + 1, v.y * nw);
    unsafeAtomicAdd(ap + 2, v.z * nw);
    unsafeAtomicAdd(ap + 3, v.w * nw);
  }
}

// ---------------- finalize: out = (agg + h*selfw + bias), optional relu ----------------

__global__ void finalize(const float* __restrict__ agg, const float* __restrict__ h,
                         const float* __restrict__ selfw, const float* __restrict__ bias,
                         float* __restrict__ out, long total, int D, int relu) {
  long idx = (long)blockIdx.x * blockDim.x + threadIdx.x;
  if (idx < total) {
    int i = (int)(idx / D);
    int j = (int)(idx - (long)i * D);
    float v = agg[idx] + h[idx] * selfw[i] + bias[j];
    if (relu) v = fmaxf(v, 0.0f);
    out[idx] = v;
  }
}

// ---------------- host orchestration ----------------

extern "C" void kernel_launch(void* const* d_in, const int* in_sizes, int n_in,
                              void* d_out, int out_size, void* d_ws, size_t ws_size,
                              hipStream_t stream) {
  const float* x  = (const float*)d_in[0];
  const float* ew = (const float*)d_in[1];
  const int*   ei = (const int*)d_in[2];
  const float* Wm[4] = {(const float*)d_in[3], (const float*)d_in[5],
                        (const float*)d_in[7], (const float*)d_in[9]};
  const float* bm[4] = {(const float*)d_in[4], (const float*)d_in[6],
                        (const float*)d_in[8], (const float*)d_in[10]};
  const int* src = ei;
  const int* dst = ei + NEDGES;

  // workspace layout (floats): norm | dinv | selfw | h | agg   (~110 MB)
  float* norm  = (float*)d_ws;
  float* dinv  = norm  + NEDGES;
  float* selfw = dinv  + NNODES;
  float* h     = selfw + NNODES;
  float* agg   = h     + (size_t)NNODES * 128;

  const int TB = 256;

  // ---- precompute gcn_norm (deg scratch lives in h buffer) ----
  zero_f32<<<1024, TB, 0, stream>>>(h, (long)NNODES);
  deg_accum<<<(NEDGES + TB - 1) / TB, TB, 0, stream>>>(dst, ew, h, NEDGES);
  dinv_selfw<<<(NNODES + TB - 1) / TB, TB, 0, stream>>>(h, dinv, selfw, NNODES);
  edge_norm<<<(NEDGES + TB - 1) / TB, TB, 0, stream>>>(src, dst, ew, dinv, norm, NEDGES);

  const int scatterBlocks = (NEDGES * 32) / TB;  // wave per edge, 8 waves/block

  // ---- layers 1-3: D=128, ReLU; activations live in agg (in-place finalize) ----
  const float* act = x;
  for (int l = 0; l < 3; ++l) {
    int tiles = (NNODES / 16) * (128 / 16);
    gemm_f32_wmma<128, 128><<<(tiles * 32) / TB, TB, 0, stream>>>(act, Wm[l], h, NNODES);
    zero_f32<<<2048, TB, 0, stream>>>(agg, (long)NNODES * 128);
    scatter_edges<128><<<scatterBlocks, TB, 0, stream>>>(h, src, dst, norm, agg, NEDGES);
    finalize<<<(int)(((long)NNODES * 128 + TB - 1) / TB), TB, 0, stream>>>(
        agg, h, selfw, bm[l], agg, (long)NNODES * 128, 128, 1);
    act = agg;
  }

  // ---- layer 4: D_out = 64, no ReLU, write d_out ----
  {
    int tiles = (NNODES / 16) * (64 / 16);
    gemm_f32_wmma<128, 64><<<(tiles * 32) / TB, TB, 0, stream>>>(act, Wm[3], h, NNODES);
    zero_f32<<<2048, TB, 0, stream>>>(agg, (long)NNODES * 64);
    scatter_edges<64><<<scatterBlocks, TB, 0, stream>>>(h, src, dst, norm, agg, NEDGES);
    finalize<<<(int)(((long)NNODES * 64 + TB - 1) / TB), TB, 0, stream>>>(
        agg, h, selfw, bm[3], (float*)d_out, (long)NNODES * 64, 64, 0);
  }
}